// SAGPoolH_57784490000620
// MI455X (gfx1250) — compile-verified
//
#include <hip/hip_runtime.h>
#include <math.h>

typedef __attribute__((ext_vector_type(2))) float v2f;
typedef __attribute__((ext_vector_type(8))) float v8f;

constexpr int BB = 128;   // batch
constexpr int NN = 512;   // nodes
constexpr int EE = 8192;  // edges
constexpr int HH = 128;   // hidden = in_dim
constexpr int OUTD = 10;

// ---------------------------------------------------------------------------
// Adjacency construction
// ---------------------------------------------------------------------------
__global__ void k_build_adj(const int* __restrict__ src, const int* __restrict__ dst,
                            float* __restrict__ A) {
  int t = blockIdx.x * blockDim.x + threadIdx.x;
  if (t >= BB * EE) return;
  int b = t / EE;
  int s = src[t];
  int d = dst[t];
  float* Ab = A + (size_t)b * NN * NN;
  Ab[(size_t)s * NN + d] = 1.0f;   // set + symmetrize (max with transpose)
  Ab[(size_t)d * NN + s] = 1.0f;
}

__global__ void k_diag_dinv(float* __restrict__ A, float* __restrict__ dinv) {
  int t = blockIdx.x * blockDim.x + threadIdx.x;
  if (t >= BB * NN) return;
  int b = t / NN, i = t % NN;
  float* row = A + (size_t)b * NN * NN + (size_t)i * NN;
  row[i] += 1.0f;                  // + I (after symmetric set; self-edge -> 2)
  float s = 0.0f;
  for (int j = 0; j < NN; ++j) s += row[j];
  dinv[t] = rsqrtf(s);
}

__global__ void k_normalize(float* __restrict__ A, const float* __restrict__ dinv) {
  unsigned t = blockIdx.x * 256u + threadIdx.x;
  unsigned total = (unsigned)BB * NN * NN;
  if (t >= total) return;
  unsigned b = t / (NN * NN);
  unsigned rem = t % (NN * NN);
  unsigned i = rem / NN, j = rem % NN;
  A[t] *= dinv[b * NN + i] * dinv[b * NN + j];
}

// ---------------------------------------------------------------------------
// Batched f32 GEMM with V_WMMA_F32_16X16X4_F32.
// C[b] = act(A[b] (MxK, row-major, ld=K) x Bm[b] (KxN, row-major, ld=N) [+bias])
// Block: 256 threads = 8 waves; block tile 64(M) x 32(N); wave tile 16x16.
// K blocked by 32, staged through LDS (stride 36 padding for bank spread).
// ---------------------------------------------------------------------------
template <bool RELU, bool BIAS>
__global__ __launch_bounds__(256) void k_gemm_wmma(
    const float* __restrict__ A, long strideA,
    const float* __restrict__ Bm, long strideB,
    float* __restrict__ C, long strideC,
    const float* __restrict__ bias, int M, int K, int N) {
  __shared__ float As[64 * 36];
  __shared__ float Bs[32 * 36];

  const float* Ab = A + (long)blockIdx.z * strideA;
  const float* Bb = Bm + (long)blockIdx.z * strideB;
  float* Cb = C + (long)blockIdx.z * strideC;

  const int m0 = blockIdx.x * 64;
  const int n0 = blockIdx.y * 32;
  const int tid = threadIdx.x;
  const int wave = tid >> 5;
  const int lane = tid & 31;
  const int hl = lane >> 4;       // which half of the wave (K split)
  const int r = lane & 15;        // row-within-16 / col-within-16
  const int wm = (wave & 3) * 16; // wave M offset in block tile
  const int wn = (wave >> 2) * 16;

  v8f acc = {0.f, 0.f, 0.f, 0.f, 0.f, 0.f, 0.f, 0.f};

  for (int kb = 0; kb < K; kb += 32) {
    // --- stage A tile: 64 x 32 (row-major, padded stride 36) ---
#pragma unroll
    for (int it = 0; it < 2; ++it) {
      int idx = tid + it * 256;           // 0..511 -> 512 float4 = 2048 floats
      int row = idx >> 3;
      int c4 = (idx & 7) << 2;
      float4 av = *(const float4*)(Ab + (long)(m0 + row) * K + kb + c4);
      *(float4*)(&As[row * 36 + c4]) = av;
    }
    // --- stage B tile: 32 x 32, stored transposed (Bs[col][k]) ---
    {
      int krow = tid >> 3;                // 0..31
      int c4 = (tid & 7) << 2;
      float4 bv = *(const float4*)(Bb + (long)(kb + krow) * N + n0 + c4);
      Bs[(c4 + 0) * 36 + krow] = bv.x;
      Bs[(c4 + 1) * 36 + krow] = bv.y;
      Bs[(c4 + 2) * 36 + krow] = bv.z;
      Bs[(c4 + 3) * 36 + krow] = bv.w;
    }
    if (kb + 32 < K) {
      __builtin_prefetch(Ab + (long)(m0 + (tid >> 2)) * K + kb + 32, 0, 0);
    }
    __syncthreads();

    // --- 8 x V_WMMA_F32_16X16X4_F32 per K-block ---
#pragma unroll
    for (int kk = 0; kk < 32; kk += 4) {
      // A frag (16x4): lanes0-15 -> M=r with K={kk,kk+1}; lanes16-31 -> K={kk+2,kk+3}
      v2f afrag = *(const v2f*)(&As[(wm + r) * 36 + kk + 2 * hl]);
      // B frag (4x16): lanes0-15 -> N=r with K={kk,kk+1}; lanes16-31 -> K={kk+2,kk+3}
      v2f bfrag = *(const v2f*)(&Bs[(wn + r) * 36 + kk + 2 * hl]);
      acc = __builtin_amdgcn_wmma_f32_16x16x4_f32(
          false, afrag, false, bfrag, (short)0, acc, false, false);
    }
    __syncthreads();
  }

  // --- epilogue: C/D layout -> VGPR v holds row (v + 8*half), col = r ---
  const int col = n0 + wn + r;
  const float bv = BIAS ? bias[col] : 0.0f;
#pragma unroll
  for (int v = 0; v < 8; ++v) {
    int row = m0 + wm + v + 8 * hl;
    float val = acc[v] + bv;
    if (RELU) val = fmaxf(val, 0.0f);
    Cb[(long)row * N + col] = val;
  }
}

// ---------------------------------------------------------------------------
// Attention score: v = Y @ aw ; s = tanh(A @ v + ab)
// ---------------------------------------------------------------------------
__global__ void k_xaw(const float* __restrict__ Y, const float* __restrict__ aw,
                      float* __restrict__ v, int n) {
  int t = blockIdx.x * blockDim.x + threadIdx.x;
  if (t >= BB * n) return;
  const float* yr = Y + (size_t)t * HH;
  float s = 0.0f;
  for (int f = 0; f < HH; ++f) s += yr[f] * aw[f];
  v[t] = s;
}

__global__ void k_ascore(const float* __restrict__ A, const float* __restrict__ v,
                         const float* __restrict__ ab, float* __restrict__ sc, int n) {
  int t = blockIdx.x * blockDim.x + threadIdx.x;
  if (t >= BB * n) return;
  int b = t / n, i = t % n;
  const float* Ar = A + (size_t)b * n * n + (size_t)i * n;
  const float* vb = v + (size_t)b * n;
  float s = 0.0f;
  for (int j = 0; j < n; ++j) s += Ar[j] * vb[j];
  sc[t] = tanhf(s + ab[0]);
}

// ---------------------------------------------------------------------------
// Top-k: full bitonic sort (descending) in LDS; block = n threads, grid = B.
// ---------------------------------------------------------------------------
__global__ void k_topk(const float* __restrict__ sc, int n, int k,
                       int* __restrict__ idx, float* __restrict__ vals) {
  extern __shared__ float smem[];
  float* skey = smem;
  int* sidx = (int*)(smem + n);
  const int b = blockIdx.x;
  const int t = threadIdx.x;
  skey[t] = sc[(size_t)b * n + t];
  sidx[t] = t;
  __syncthreads();
  for (int kk = 2; kk <= n; kk <<= 1) {
    for (int j = kk >> 1; j > 0; j >>= 1) {
      int ixj = t ^ j;
      if (ixj > t) {
        float a = skey[t], c = skey[ixj];
        bool desc = ((t & kk) == 0);
        bool sw = desc ? (a < c) : (a > c);
        if (sw) {
          skey[t] = c; skey[ixj] = a;
          int tmp = sidx[t]; sidx[t] = sidx[ixj]; sidx[ixj] = tmp;
        }
      }
      __syncthreads();
    }
  }
  if (t < k) {
    idx[(size_t)b * k + t] = sidx[t];
    vals[(size_t)b * k + t] = skey[t];
  }
}

// ---------------------------------------------------------------------------
// Pool gathers + readout accumulate
// ---------------------------------------------------------------------------
__global__ void k_gather_x(const float* __restrict__ Y, const int* __restrict__ idx,
                           const float* __restrict__ vals, float* __restrict__ Xp,
                           int n, int k) {
  int t = blockIdx.x * blockDim.x + threadIdx.x;
  if (t >= BB * k * HH) return;
  int b = t / (k * HH);
  int rem = t % (k * HH);
  int i = rem / HH, f = rem % HH;
  int si = idx[(size_t)b * k + i];
  Xp[t] = Y[((size_t)b * n + si) * HH + f] * vals[(size_t)b * k + i];
}

__global__ void k_gather_a(const float* __restrict__ A, const int* __restrict__ idx,
                           float* __restrict__ An, int n, int k) {
  int t = blockIdx.x * blockDim.x + threadIdx.x;
  if (t >= BB * k * k) return;
  int b = t / (k * k);
  int rem = t % (k * k);
  int i = rem / k, j = rem % k;
  const int* ib = idx + (size_t)b * k;
  An[t] = A[(size_t)b * n * n + (size_t)ib[i] * n + ib[j]];
}

__global__ void k_readout(const float* __restrict__ Xp, float* __restrict__ r, int k) {
  int t = blockIdx.x * blockDim.x + threadIdx.x;
  if (t >= BB * HH) return;
  int b = t / HH, f = t % HH;
  const float* xb = Xp + (size_t)b * k * HH + f;
  float s = 0.0f, m = -INFINITY;
  for (int i = 0; i < k; ++i) {
    float x = xb[(size_t)i * HH];
    s += x;
    m = fmaxf(m, x);
  }
  r[(size_t)b * 256 + f] += s / (float)k;
  r[(size_t)b * 256 + 128 + f] += m;
}

// ---------------------------------------------------------------------------
// Final MLP (tiny)
// ---------------------------------------------------------------------------
__global__ void k_mlp1(const float* __restrict__ r, const float* __restrict__ w,
                       const float* __restrict__ bias, float* __restrict__ h) {
  int t = blockIdx.x * blockDim.x + threadIdx.x;
  if (t >= BB * HH) return;
  int b = t / HH, j = t % HH;
  float s = bias[j];
  for (int f = 0; f < 256; ++f) s += r[(size_t)b * 256 + f] * w[(size_t)f * HH + j];
  h[t] = fmaxf(s, 0.0f);
}

__global__ void k_mlp2(const float* __restrict__ h1, const float* __restrict__ w,
                       const float* __restrict__ bias, float* __restrict__ h2) {
  int t = blockIdx.x * blockDim.x + threadIdx.x;
  if (t >= BB * 64) return;
  int b = t / 64, j = t % 64;
  float s = bias[j];
  for (int f = 0; f < HH; ++f) s += h1[(size_t)b * HH + f] * w[(size_t)f * 64 + j];
  h2[t] = fmaxf(s, 0.0f);
}

__global__ void k_mlp3(const float* __restrict__ h2, const float* __restrict__ w,
                       const float* __restrict__ bias, float* __restrict__ out) {
  int t = blockIdx.x * blockDim.x + threadIdx.x;
  if (t >= BB * OUTD) return;
  int b = t / OUTD, j = t % OUTD;
  float s = bias[j];
  for (int f = 0; f < 64; ++f) s += h2[(size_t)b * 64 + f] * w[(size_t)f * OUTD + j];
  out[t] = s;
}

// ---------------------------------------------------------------------------
static inline int cdiv(int a, int b) { return (a + b - 1) / b; }

extern "C" void kernel_launch(void* const* d_in, const int* in_sizes, int n_in,
                              void* d_out, int out_size, void* d_ws, size_t ws_size,
                              hipStream_t stream) {
  const float* x    = (const float*)d_in[0];
  const int* esrc   = (const int*)d_in[1];
  const int* edst   = (const int*)d_in[2];
  const float* W[3]  = {(const float*)d_in[3], (const float*)d_in[7], (const float*)d_in[11]};
  const float* bi[3] = {(const float*)d_in[4], (const float*)d_in[8], (const float*)d_in[12]};
  const float* aw[3] = {(const float*)d_in[5], (const float*)d_in[9], (const float*)d_in[13]};
  const float* ab[3] = {(const float*)d_in[6], (const float*)d_in[10], (const float*)d_in[14]};
  const float* mw1 = (const float*)d_in[15];
  const float* mb1 = (const float*)d_in[16];
  const float* mw2 = (const float*)d_in[17];
  const float* mb2 = (const float*)d_in[18];
  const float* mw3 = (const float*)d_in[19];
  const float* mb3 = (const float*)d_in[20];
  float* out = (float*)d_out;

  // --- workspace carve-up ---
  char* ws = (char*)d_ws;
  size_t off = 0;
  auto allocf = [&](size_t nfl) {
    float* p = (float*)(ws + off);
    off += nfl * sizeof(float);
    return p;
  };
  float* A0  = allocf((size_t)BB * NN * NN);        // 512x512 adjacency
  float* Yb  = allocf((size_t)BB * NN * HH);        // GCN output (max n)
  float* XW  = allocf((size_t)BB * NN * HH);        // X@W temp (max n)
  float* Xp1 = allocf((size_t)BB * 256 * HH);
  float* A1  = allocf((size_t)BB * 256 * 256);
  float* Xp2 = allocf((size_t)BB * 128 * HH);
  float* A2  = allocf((size_t)BB * 128 * 128);
  float* Xp3 = allocf((size_t)BB * 64 * HH);
  float* vbuf = allocf((size_t)BB * NN);
  float* sc   = allocf((size_t)BB * NN);
  float* vals = allocf((size_t)BB * NN);
  int* idx    = (int*)allocf((size_t)BB * NN);
  float* dinv = allocf((size_t)BB * NN);
  float* rbuf = allocf((size_t)BB * 256);
  float* h1   = allocf((size_t)BB * HH);
  float* h2   = allocf((size_t)BB * 64);
  (void)ws_size; (void)n_in; (void)in_sizes; (void)out_size;

  // --- normalized adjacency ---
  hipMemsetAsync(A0, 0, (size_t)BB * NN * NN * sizeof(float), stream);
  hipMemsetAsync(rbuf, 0, (size_t)BB * 256 * sizeof(float), stream);
  k_build_adj<<<cdiv(BB * EE, 256), 256, 0, stream>>>(esrc, edst, A0);
  k_diag_dinv<<<cdiv(BB * NN, 256), 256, 0, stream>>>(A0, dinv);
  {
    unsigned total = (unsigned)BB * NN * NN;
    k_normalize<<<(total + 255) / 256, 256, 0, stream>>>(A0, dinv);
  }

  // --- three GCN + SAGPool levels ---
  const float* Xin = x;
  float* Acur = A0;
  float* Xps[3] = {Xp1, Xp2, Xp3};
  float* Anexts[3] = {A1, A2, nullptr};
  int ns[3] = {512, 256, 128};
  int ks[3] = {256, 128, 64};

  for (int L = 0; L < 3; ++L) {
    int n = ns[L], k = ks[L];
    // XW = Xin @ W  (M=n, K=HH, N=HH)
    k_gemm_wmma<false, false><<<dim3(n / 64, HH / 32, BB), 256, 0, stream>>>(
        Xin, (long)n * HH, W[L], 0L, XW, (long)n * HH, nullptr, n, HH, HH);
    // Y = relu(A @ XW + bias)  (M=n, K=n, N=HH)
    k_gemm_wmma<true, true><<<dim3(n / 64, HH / 32, BB), 256, 0, stream>>>(
        Acur, (long)n * n, XW, (long)n * HH, Yb, (long)n * HH, bi[L], n, n, HH);
    // score = tanh(A @ (Y @ aw) + ab)
    k_xaw<<<cdiv(BB * n, 256), 256, 0, stream>>>(Yb, aw[L], vbuf, n);
    k_ascore<<<cdiv(BB * n, 256), 256, 0, stream>>>(Acur, vbuf, ab[L], sc, n);
    // top-k (bitonic, one block per batch)
    k_topk<<<BB, n, (size_t)n * 8, stream>>>(sc, n, k, idx, vals);
    // gather pooled X and A
    k_gather_x<<<cdiv(BB * k * HH, 256), 256, 0, stream>>>(Yb, idx, vals, Xps[L], n, k);
    if (Anexts[L]) {
      k_gather_a<<<cdiv(BB * k * k, 256), 256, 0, stream>>>(Acur, idx, Anexts[L], n, k);
    }
    // readout accumulate
    k_readout<<<cdiv(BB * HH, 256), 256, 0, stream>>>(Xps[L], rbuf, k);
    Xin = Xps[L];
    Acur = Anexts[L];
  }

  // --- MLP head ---
  k_mlp1<<<cdiv(BB * HH, 256), 256, 0, stream>>>(rbuf, mw1, mb1, h1);
  k_mlp2<<<cdiv(BB * 64, 256), 256, 0, stream>>>(h1, mw2, mb2, h2);
  k_mlp3<<<cdiv(BB * OUTD, 64), 64, 0, stream>>>(h2, mw3, mb3, out);
}